// MADGraph_4612794876397
// MI455X (gfx1250) — compile-verified
//
#include <hip/hip_runtime.h>
#include <math.h>

// MADGraph scoring kernel for MI455X (gfx1250, wave32).
// One wave per query. FP32 WMMA (v_wmma_f32_16x16x4_f32) computes both the
// 16 logit dot-products (A x B) and the 16 squared distances (A x A^T diag)
// in full f32 precision. Tables are L2-resident; edge gathers go to HBM.

typedef __attribute__((ext_vector_type(2))) float v2f;
typedef __attribute__((ext_vector_type(8))) float v8f;

#define N_NODES_C 10000
#define DFEAT     128
#define NSAMP     8
#define NQ        65536

__global__ __launch_bounds__(256) void madgraph_wmma_kernel(
    const int*   __restrict__ src,
    const int*   __restrict__ dst,
    const int*   __restrict__ mid0,
    const int*   __restrict__ mid1,
    const float* __restrict__ position,
    const float* __restrict__ src_field,
    const float* __restrict__ dst_field,
    const float* __restrict__ uncertainty,
    const float* __restrict__ edge,
    float*       __restrict__ out)
{
    const int lane = threadIdx.x & 31;
    const int q    = blockIdx.x * 8 + (threadIdx.x >> 5);   // 8 waves / block

    const int sn = src[q];
    const int dn = dst[q];

    // A-matrix row m (0..15): rows 0..7 = srcdiff[s], rows 8..15 = dstdiff[s]
    // B-matrix col j = m     : cols 0..7 = dst_field[dst], 8..15 = src_field[src]
    const int  m       = lane & 15;
    const int  koff    = (lane >> 4) << 1;    // lanes 0-15 -> K 0,1 ; lanes 16-31 -> K 2,3
    const int  s       = m & 7;
    const bool lowHalf = (m < 8);

    const int midrow = lowHalf ? mid0[q * NSAMP + s] : mid1[q * NSAMP + s];
    const int ancrow = lowHalf ? sn : dn;

    const float* aptr = position + (size_t)ancrow * DFEAT + koff;
    const float* mptr = position + (size_t)midrow * DFEAT + koff;
    const float* fptr = (lowHalf ? (dst_field + (size_t)dn * DFEAT)
                                 : (src_field + (size_t)sn * DFEAT)) + koff;

    // Edge-table gather (HBM, random) issued early to overlap with WMMA loop.
    float ev = 0.0f;
    if (lane < 16) {
        const int row = lowHalf ? midrow : sn;
        const int col = lowHalf ? dn     : midrow;
        ev = edge[(size_t)row * N_NODES_C + col];
    }

    v8f accL = {0.f, 0.f, 0.f, 0.f, 0.f, 0.f, 0.f, 0.f};   // logits = A x B
    v8f accG = {0.f, 0.f, 0.f, 0.f, 0.f, 0.f, 0.f, 0.f};   // gram   = A x A^T

    #pragma unroll 4
    for (int k = 0; k < DFEAT; k += 4) {
        const float2 a  = *(const float2*)(aptr + k);
        const float2 mm = *(const float2*)(mptr + k);
        const float2 f  = *(const float2*)(fptr + k);
        v2f av; av.x = a.x - mm.x; av.y = a.y - mm.y;
        v2f bv; bv.x = f.x;        bv.y = f.y;
        // 8 args: (neg_a, A, neg_b, B, c_mod, C, reuse_a, reuse_b)
        accL = __builtin_amdgcn_wmma_f32_16x16x4_f32(false, av, false, bv,
                                                     (short)0, accL, false, false);
        accG = __builtin_amdgcn_wmma_f32_16x16x4_f32(false, av, false, av,
                                                     (short)0, accG, false, false);
    }

    // Diagonal extraction. C/D layout: VGPR r -> (M=r, N=lane) for lanes 0-15,
    // (M=r+8, N=lane-16) for lanes 16-31. diag[r] @ lane r, diag[r+8] @ lane r+24.
    float myLogit = 0.0f, myGram = 0.0f;
    #pragma unroll
    for (int r = 0; r < 8; ++r) {
        const float l_lo = __shfl(accL[r], r,      32);
        const float l_hi = __shfl(accL[r], r + 24, 32);
        const float g_lo = __shfl(accG[r], r,      32);
        const float g_hi = __shfl(accG[r], r + 24, 32);
        if (lane == r)     { myLogit = l_lo; myGram = g_lo; }
        if (lane == r + 8) { myLogit = l_hi; myGram = g_hi; }
    }

    const float u = uncertainty[0];
    float logit, score;
    if (lane < 16) {
        logit = myLogit + u * ev;
        score = 1.0f - sqrtf(myGram);
    } else {
        logit = 0.0f;
        score = -INFINITY;
    }

    // softmax over the 16 valid lanes (butterfly across all 32 lanes)
    float mmax = score;
    #pragma unroll
    for (int off = 16; off >= 1; off >>= 1)
        mmax = fmaxf(mmax, __shfl_xor(mmax, off, 32));

    const float e = (lane < 16) ? expf(score - mmax) : 0.0f;

    float esum = e;
    #pragma unroll
    for (int off = 16; off >= 1; off >>= 1)
        esum += __shfl_xor(esum, off, 32);

    float contrib = logit * e;
    #pragma unroll
    for (int off = 16; off >= 1; off >>= 1)
        contrib += __shfl_xor(contrib, off, 32);

    if (lane == 0)
        out[q] = contrib / esum;
}

extern "C" void kernel_launch(void* const* d_in, const int* in_sizes, int n_in,
                              void* d_out, int out_size, void* d_ws, size_t ws_size,
                              hipStream_t stream) {
    const int*   src         = (const int*)  d_in[0];
    const int*   dst         = (const int*)  d_in[1];
    const int*   mid0        = (const int*)  d_in[2];
    const int*   mid1        = (const int*)  d_in[3];
    const float* position    = (const float*)d_in[4];
    const float* src_field   = (const float*)d_in[5];
    const float* dst_field   = (const float*)d_in[6];
    const float* uncertainty = (const float*)d_in[7];
    const float* edge        = (const float*)d_in[8];
    float*       out         = (float*)d_out;

    // 1 wave per query, 8 waves (256 threads) per block
    madgraph_wmma_kernel<<<NQ / 8, 256, 0, stream>>>(
        src, dst, mid0, mid1, position, src_field, dst_field,
        uncertainty, edge, out);
}